// FCOS_58188216926705
// MI455X (gfx1250) — compile-verified
//
#include <hip/hip_runtime.h>
#include <hip/hip_bf16.h>
#include <stdint.h>
#include <limits.h>

#define NBOX   8192
#define NWORDS (NBOX / 64)   /* 128 uint64 words per mask row */
#define IOU_THR 0.5f

typedef unsigned int       u32;
typedef unsigned long long u64;
typedef u32 uint4v __attribute__((ext_vector_type(4)));
typedef int int8v  __attribute__((ext_vector_type(8)));
typedef int int4v  __attribute__((ext_vector_type(4)));

/* ---------------- init: maxval = INT_MIN (as int bits; final max is positive) ---- */
__global__ void FCOS_init_kernel(int* maxint) { *maxint = INT_MIN; }

/* ---------------- decode boxes + centerness + global max reduction --------------- */
__global__ void __launch_bounds__(256)
FCOS_decode_kernel(const float* __restrict__ deltas,
                   const float* __restrict__ loc,
                   const float* __restrict__ gtb,
                   const int*   __restrict__ stride_p,
                   float* __restrict__ boxes,
                   float* __restrict__ ctr,
                   int*   __restrict__ maxint) {
  int i = blockIdx.x * blockDim.x + threadIdx.x;
  float s  = (float)stride_p[0];
  float xc = loc[2 * i + 0], yc = loc[2 * i + 1];
  float d0 = fmaxf(deltas[4 * i + 0], 0.0f);
  float d1 = fmaxf(deltas[4 * i + 1], 0.0f);
  float d2 = fmaxf(deltas[4 * i + 2], 0.0f);
  float d3 = fmaxf(deltas[4 * i + 3], 0.0f);
  float x1 = xc - d0 * s, y1 = yc - d1 * s;
  float x2 = xc + d2 * s, y2 = yc + d3 * s;
  boxes[4 * i + 0] = x1; boxes[4 * i + 1] = y1;
  boxes[4 * i + 2] = x2; boxes[4 * i + 3] = y2;

  float l = (xc - gtb[4 * i + 0]) / s;
  float t = (yc - gtb[4 * i + 1]) / s;
  float r = (gtb[4 * i + 2] - xc) / s;
  float b = (gtb[4 * i + 3] - yc) / s;
  ctr[i] = sqrtf((fminf(l, r) * fminf(t, b)) / (fmaxf(l, r) * fmaxf(t, b)));

  __shared__ float red[256];
  red[threadIdx.x] = fmaxf(fmaxf(x1, y1), fmaxf(x2, y2));
  __syncthreads();
  for (int off = 128; off > 0; off >>= 1) {
    if ((int)threadIdx.x < off)
      red[threadIdx.x] = fmaxf(red[threadIdx.x], red[threadIdx.x + off]);
    __syncthreads();
  }
  if (threadIdx.x == 0) atomicMax(maxint, __float_as_int(red[0]));
}

/* ---------------- class-aware offset: boxes_nms = boxes + cls*(max+1) ------------ */
__global__ void __launch_bounds__(256)
FCOS_offset_kernel(const int* __restrict__ class_ids,
                   const int* __restrict__ maxint,
                   const float* __restrict__ boxes,
                   float* __restrict__ boxesNms) {
  int i = blockIdx.x * blockDim.x + threadIdx.x;
  float maxv = __int_as_float(*maxint);
  float off  = (float)class_ids[i] * (maxv + 1.0f);
#pragma unroll
  for (int c = 0; c < 4; ++c) boxesNms[4 * i + c] = boxes[4 * i + c] + off;
}

/* ---------------- exact stable descending rank (== argsort(-scores)) ------------- */
__global__ void __launch_bounds__(256)
FCOS_rank_kernel(const float* __restrict__ scores,
                 const float* __restrict__ boxesNms,
                 int*   __restrict__ order,
                 float* __restrict__ bs) {
  int i = blockIdx.x * blockDim.x + threadIdx.x;
  float si = scores[i];
  int rank = 0;
  __shared__ float sc[256];
  for (int base = 0; base < NBOX; base += 256) {
    sc[threadIdx.x] = scores[base + threadIdx.x];
    __syncthreads();
#pragma unroll 8
    for (int jj = 0; jj < 256; ++jj) {
      int j = base + jj;
      float sj = sc[jj];
      rank += (sj > si) || (sj == si && j < i);
    }
    __syncthreads();
  }
  order[rank] = i;
#pragma unroll
  for (int c = 0; c < 4; ++c) bs[4 * rank + c] = boxesNms[4 * i + c];
}

/* ---------------- N x N IoU bitmask; column tile staged in LDS via TDM ----------- */
__global__ void __launch_bounds__(64)
FCOS_mask_kernel(const float* __restrict__ bs, u64* __restrict__ mask) {
  __shared__ float colb[64 * 4];   /* 64 column boxes, 1 KB */

  if (threadIdx.x < 32) {          /* one wave32 issues the TDM DMA */
    u64 gaddr  = (u64)(uintptr_t)(bs + (size_t)blockIdx.x * 256);
    u32 ldsoff = (u32)(uintptr_t)(&colb[0]);
#if __has_builtin(__builtin_amdgcn_tensor_load_to_lds) && __has_builtin(__builtin_amdgcn_s_wait_tensorcnt)
    /* D# group 0: count=1, lds_addr, global_addr[56:0], type=2 */
    uint4v g0;
    g0[0] = 1u;
    g0[1] = ldsoff;
    g0[2] = (u32)gaddr;
    g0[3] = (u32)((gaddr >> 32) & 0x01ffffffull) | 0x80000000u;
    /* D# group 1: data_size=4B; tensor_dim0=32768; tensor_dim1=1; tile_dim0=256 */
    int8v g1;
    g1[0] = 0x00020000;          /* data_size = 2 -> 4 bytes            */
    g1[1] = (int)0x80000000u;    /* tensor_dim0[15:0]=0x8000 at bit 48  */
    g1[2] = 0x00010000;          /* tensor_dim1 = 1 at bit 80           */
    g1[3] = 0x01000000;          /* tile_dim0 = 256 at bit 112          */
    g1[4] = 0;                   /* tile_dim1 = tile_dim2 = 0 (unused)  */
    g1[5] = 32768;               /* tensor_dim0_stride[31:0]            */
    g1[6] = 0;
    g1[7] = 0;
    int4v gz4; gz4[0] = 0; gz4[1] = 0; gz4[2] = 0; gz4[3] = 0;
    int8v gz8; gz8[0] = 0; gz8[1] = 0; gz8[2] = 0; gz8[3] = 0;
    gz8[4] = 0; gz8[5] = 0; gz8[6] = 0; gz8[7] = 0;
    /* 6-arg variant (clang-23 / therock headers):
       (uint32x4 g0, int32x8 g1, int32x4 g2, int32x4 g3, int32x8 extra, i32 cpol) */
    __builtin_amdgcn_tensor_load_to_lds(g0, g1, gz4, gz4, gz8, 0);
    __builtin_amdgcn_s_wait_tensorcnt(0);
#else
    for (int k = threadIdx.x; k < 256; k += 32)
      colb[k] = bs[(size_t)blockIdx.x * 256 + k];
#endif
  }
  __syncthreads();

  int i = blockIdx.y * 64 + threadIdx.x;       /* this thread's row */
  float x1 = bs[4 * i + 0], y1 = bs[4 * i + 1];
  float x2 = bs[4 * i + 2], y2 = bs[4 * i + 3];
  float areaI = (x2 - x1) * (y2 - y1);

  u64 bits = 0ull;
#pragma unroll 4
  for (int b = 0; b < 64; ++b) {
    float bx1 = colb[4 * b + 0], by1 = colb[4 * b + 1];
    float bx2 = colb[4 * b + 2], by2 = colb[4 * b + 3];
    float areaB = (bx2 - bx1) * (by2 - by1);
    float iw = fmaxf(fminf(x2, bx2) - fmaxf(x1, bx1), 0.0f);
    float ih = fmaxf(fminf(y2, by2) - fmaxf(y1, by1), 0.0f);
    float inter = iw * ih;
    float iou = inter / (areaI + areaB - inter);
    bits |= (u64)(iou > IOU_THR) << b;
  }
  mask[(size_t)i * NWORDS + blockIdx.x] = bits;
}

/* ---------------- serial greedy scan: one workgroup, OR 128 words per kept row --- */
__global__ void __launch_bounds__(128)
FCOS_scan_kernel(const u64* __restrict__ mask,
                 const int* __restrict__ order,
                 float* __restrict__ kf) {
  __shared__ u64 sup[NWORDS];
  sup[threadIdx.x] = 0ull;
  __syncthreads();
  for (int i = 0; i < NBOX; ++i) {
    bool kept = ((sup[i >> 6] >> (i & 63)) & 1ull) == 0ull;
    __syncthreads();                                   /* reads before OR writes */
    if (kept) sup[threadIdx.x] |= mask[(size_t)i * NWORDS + threadIdx.x];
    if (threadIdx.x == 0) kf[order[i]] = kept ? 1.0f : 0.0f;
    __syncthreads();
  }
}

/* ---------------- output assembly: [boxes*kf, ctr, kf] --------------------------- */
__global__ void __launch_bounds__(256)
FCOS_finalize_kernel(const float* __restrict__ boxes,
                     const float* __restrict__ ctr,
                     const float* __restrict__ kf,
                     float* __restrict__ out) {
  int i = blockIdx.x * blockDim.x + threadIdx.x;
  float k = kf[i];
#pragma unroll
  for (int c = 0; c < 4; ++c) out[6 * i + c] = boxes[4 * i + c] * k;
  out[6 * i + 4] = ctr[i];
  out[6 * i + 5] = k;
}

extern "C" void kernel_launch(void* const* d_in, const int* in_sizes, int n_in,
                              void* d_out, int out_size, void* d_ws, size_t ws_size,
                              hipStream_t stream) {
  const float* deltas    = (const float*)d_in[0];
  const float* locations = (const float*)d_in[1];
  const float* scores    = (const float*)d_in[2];
  const int*   class_ids = (const int*)  d_in[3];
  const float* gt_boxes  = (const float*)d_in[4];
  const int*   stride_p  = (const int*)  d_in[5];

  char* ws = (char*)d_ws;
  float* boxes    = (float*)(ws + 0);        /* N*4 f32 = 128 KB */
  float* boxesNms = (float*)(ws + 131072);   /* N*4 f32          */
  float* bs       = (float*)(ws + 262144);   /* sorted boxes     */
  float* ctr      = (float*)(ws + 393216);   /* N f32            */
  int*   order    = (int*)  (ws + 425984);   /* N i32            */
  float* kf       = (float*)(ws + 458752);   /* N f32            */
  int*   maxint   = (int*)  (ws + 491520);   /* 1 i32            */
  u64*   mask     = (u64*)  (ws + 524288);   /* N*128 u64 = 8 MB */

  dim3 b256(256), g32(NBOX / 256);
  FCOS_init_kernel<<<1, 1, 0, stream>>>(maxint);
  FCOS_decode_kernel<<<g32, b256, 0, stream>>>(deltas, locations, gt_boxes,
                                               stride_p, boxes, ctr, maxint);
  FCOS_offset_kernel<<<g32, b256, 0, stream>>>(class_ids, maxint, boxes, boxesNms);
  FCOS_rank_kernel<<<g32, b256, 0, stream>>>(scores, boxesNms, order, bs);
  FCOS_mask_kernel<<<dim3(NWORDS, NWORDS), dim3(64), 0, stream>>>(bs, mask);
  FCOS_scan_kernel<<<1, 128, 0, stream>>>(mask, order, kf);
  FCOS_finalize_kernel<<<g32, b256, 0, stream>>>(boxes, ctr, kf, (float*)d_out);
}